// GConv_51213190038088
// MI455X (gfx1250) — compile-verified
//
#include <hip/hip_runtime.h>

typedef __attribute__((ext_vector_type(2))) float v2f;
typedef __attribute__((ext_vector_type(8))) float v8f;

#define HID 128

// ---------------------------------------------------------------- utilities
__global__ void fill0_kernel(float* __restrict__ p, long n) {
    long i = (long)blockIdx.x * blockDim.x + threadIdx.x;
    if (i < n) p[i] = 0.0f;
}

// ------------------------------------------------------------- degree accum
__global__ void deg_kernel(const int* __restrict__ col, const float* __restrict__ w,
                           float* __restrict__ deg, int E) {
    int e = blockIdx.x * blockDim.x + threadIdx.x;
    if (e < E) atomicAdd(&deg[col[e]], w[e]);
}

// deg += 1 (self loop), dis = rsqrt(deg) in place
__global__ void rsqrt_kernel(float* __restrict__ deg, int N) {
    int i = blockIdx.x * blockDim.x + threadIdx.x;
    if (i < N) deg[i] = rsqrtf(deg[i] + 1.0f);
}

__global__ void norm_kernel(const int* __restrict__ row, const int* __restrict__ col,
                            const float* __restrict__ w, const float* __restrict__ dis,
                            float* __restrict__ norm, int E) {
    int e = blockIdx.x * blockDim.x + threadIdx.x;
    if (e < E) norm[e] = dis[row[e]] * w[e] * dis[col[e]];
}

// ------------------------------------------------------------- WMMA GEMM
// C[N x 128] = A[N x 128] * W[128 x 128], f32 via V_WMMA_F32_16X16X4_F32.
// Block: 256 threads = 8 waves; each wave computes a 16x128 output strip.
// W is staged in LDS *K-pair interleaved*: element W[k][n] lives at
//   Wl[(k>>1)*256 + n*2 + (k&1)]
// so a B fragment {W[k][n], W[k+1][n]} is one contiguous ds_load_b64.
__global__ __launch_bounds__(256) void gemm128_wmma_kernel(
    const float* __restrict__ A, const float* __restrict__ W,
    float* __restrict__ C, int N) {
    __shared__ float Wl[HID * HID];

    // cooperative interleaving load: p indexes 2x2 element blocks
    // (k-pair kp = 0..63, column-pair np = 0..63)
    {
        const float2* Wg2 = (const float2*)W;   // 64 float2 per row
        float4* Wl4 = (float4*)Wl;
        #pragma unroll
        for (int it = 0; it < 4096 / 256; ++it) {
            int p  = threadIdx.x + it * 256;    // 0..4095
            int kp = p >> 6;                    // which k-pair
            int np = p & 63;                    // which column-pair
            float2 lo = Wg2[(2 * kp) * 64 + np];      // W[2kp  ][2np..2np+1]
            float2 hi = Wg2[(2 * kp + 1) * 64 + np];  // W[2kp+1][2np..2np+1]
            float4 v; v.x = lo.x; v.y = hi.x; v.z = lo.y; v.w = hi.y;
            Wl4[kp * 64 + np] = v;              // float addr: kp*256 + np*4
        }
    }
    __syncthreads();

    const int wave = threadIdx.x >> 5;
    const int lane = threadIdx.x & 31;
    const int half = lane >> 4;     // K sub-half select
    const int m    = lane & 15;     // row (A) / col (B,C)

    const long row0 = (long)blockIdx.x * 128 + wave * 16;

    // clamp A row for loads so EXEC stays all-ones through the WMMA loop
    long ra = row0 + m;
    if (ra > (long)N - 1) ra = (long)N - 1;
    const float* __restrict__ arow = A + ra * HID;

    v8f acc[8] = {};

    for (int k0 = 0; k0 < HID; k0 += 4) {
        // A frag (16x4): lanes 0-15 hold K=k0+0,k0+1 ; lanes 16-31 hold K=k0+2,k0+3
        v2f a;
        a.x = arow[k0 + 2 * half];
        a.y = arow[k0 + 2 * half + 1];
        const int kp = (k0 >> 1) + half;        // k-pair index for this half-wave
        const float* __restrict__ brow = Wl + kp * 256;
        #pragma unroll
        for (int t = 0; t < 8; ++t) {
            const int col0 = t * 16;
            // contiguous pair {W[k][n], W[k+1][n]} -> single ds_load_b64
            v2f b = *(const v2f*)(brow + (col0 + m) * 2);
            acc[t] = __builtin_amdgcn_wmma_f32_16x16x4_f32(
                false, a, false, b, (short)0, acc[t], false, false);
        }
    }

    // C/D layout: VGPR r -> row (row0 + r + 8*half), col (col0 + m)
    #pragma unroll
    for (int t = 0; t < 8; ++t) {
        const int col0 = t * 16;
        #pragma unroll
        for (int r = 0; r < 8; ++r) {
            long rr = row0 + r + 8 * half;
            if (rr < N) C[rr * HID + col0 + m] = acc[t][r];
        }
    }
}

// --------------------------------------------------- self-loop init of agg
// agg[i, :] = dis[i]^2 * xw[i, :]
__global__ void init_agg_kernel(const float* __restrict__ dis, const float* __restrict__ xw,
                                float* __restrict__ agg, int N) {
    long i = (long)blockIdx.x * blockDim.x + threadIdx.x;   // over N*32 float4 chunks
    if (i >= (long)N * 32) return;
    int node = (int)(i >> 5);
    float s = dis[node];
    s = s * s;
    float4 v = ((const float4*)xw)[i];
    float4 r;
    r.x = s * v.x; r.y = s * v.y; r.z = s * v.z; r.w = s * v.w;
    ((float4*)agg)[i] = r;
}

// ------------------------------------------------------------ edge scatter
// one wave per edge: lane j handles channels 4j..4j+3
__global__ __launch_bounds__(256) void scatter_kernel(
    const int* __restrict__ row, const int* __restrict__ col,
    const float* __restrict__ norm, const float* __restrict__ xw,
    float* __restrict__ agg, int E) {
    int e = blockIdx.x * 8 + (threadIdx.x >> 5);
    if (e >= E) return;
    int lane = threadIdx.x & 31;
    int r = row[e];
    int c = col[e];
    float wv = norm[e];
    float4 v = ((const float4*)(xw + (long)r * HID))[lane];
    float* dst = agg + (long)c * HID + lane * 4;
    atomicAdd(dst + 0, wv * v.x);
    atomicAdd(dst + 1, wv * v.y);
    atomicAdd(dst + 2, wv * v.z);
    atomicAdd(dst + 3, wv * v.w);
}

// ------------------------------------- bias + PReLU + pooled-readout accum
__global__ void finalize_kernel(const float* __restrict__ agg, const float* __restrict__ bias,
                                const float* __restrict__ alpha, const int* __restrict__ batch,
                                float* __restrict__ zout, float* __restrict__ g,
                                int goff, int N) {
    long i = (long)blockIdx.x * blockDim.x + threadIdx.x;   // over N*32 float4 chunks
    if (i >= (long)N * 32) return;
    int node = (int)(i >> 5);
    int j = (int)(i & 31);
    float4 v = ((const float4*)agg)[i];
    float4 bb = ((const float4*)bias)[j];
    float4 aa = ((const float4*)alpha)[j];
    float4 z;
    z.x = v.x + bb.x; z.x = z.x > 0.0f ? z.x : aa.x * z.x;
    z.y = v.y + bb.y; z.y = z.y > 0.0f ? z.y : aa.y * z.y;
    z.z = v.z + bb.z; z.z = z.z > 0.0f ? z.z : aa.z * z.z;
    z.w = v.w + bb.w; z.w = z.w > 0.0f ? z.w : aa.w * z.w;
    ((float4*)zout)[i] = z;
    float* gd = g + (long)batch[node] * (2 * HID) + goff + j * 4;
    atomicAdd(gd + 0, z.x);
    atomicAdd(gd + 1, z.y);
    atomicAdd(gd + 2, z.z);
    atomicAdd(gd + 3, z.w);
}

// ------------------------------------------------------------------- host
static inline int cdiv(long a, long b) { return (int)((a + b - 1) / b); }

extern "C" void kernel_launch(void* const* d_in, const int* in_sizes, int n_in,
                              void* d_out, int out_size, void* d_ws, size_t ws_size,
                              hipStream_t stream) {
    const int*   batch = (const int*)d_in[0];
    const float* x     = (const float*)d_in[1];
    const int*   eidx  = (const int*)d_in[2];
    const float* ew    = (const float*)d_in[3];
    const float* W1    = (const float*)d_in[4];
    const float* b1    = (const float*)d_in[5];
    const float* W2    = (const float*)d_in[6];
    const float* b2    = (const float*)d_in[7];
    const float* pa    = (const float*)d_in[8];

    const int N = in_sizes[0];
    const int E = in_sizes[3];
    const int* erow = eidx;
    const int* ecol = eidx + E;

    // workspace layout (16B-aligned chunks)
    float* ws   = (float*)d_ws;
    float* dis  = ws;                                   // [N]   (deg, then rsqrt in place)
    float* norm = dis + (((long)N + 3) & ~3L);          // [E]
    float* xw   = norm + (((long)E + 3) & ~3L);         // [N*128]
    float* z1   = xw + (long)N * HID;                   // [N*128]

    float* out = (float*)d_out;
    float* z2  = out;                                   // [N*128]
    float* g   = out + (long)N * HID;                   // [64*256]
    float* agg = z2;                                    // reuse z2 region as accumulator
    const long gsz = (long)out_size - (long)N * HID;

    const int B = 256;

    // ---- normalization factors
    fill0_kernel<<<cdiv(N, B), B, 0, stream>>>(dis, N);
    fill0_kernel<<<cdiv(gsz, B), B, 0, stream>>>(g, gsz);
    deg_kernel<<<cdiv(E, B), B, 0, stream>>>(ecol, ew, dis, E);
    rsqrt_kernel<<<cdiv(N, B), B, 0, stream>>>(dis, N);
    norm_kernel<<<cdiv(E, B), B, 0, stream>>>(erow, ecol, ew, dis, norm, E);

    const int gemm_blocks = cdiv(N, 128);
    const long nchunks = (long)N * 32;

    // ---- layer 1: xw = x @ W1 ; agg = selfloop + scatter ; z1 = prelu(agg + b1)
    gemm128_wmma_kernel<<<gemm_blocks, 256, 0, stream>>>(x, W1, xw, N);
    init_agg_kernel<<<cdiv(nchunks, B), B, 0, stream>>>(dis, xw, agg, N);
    scatter_kernel<<<cdiv(E, 8), 256, 0, stream>>>(erow, ecol, norm, xw, agg, E);
    finalize_kernel<<<cdiv(nchunks, B), B, 0, stream>>>(agg, b1, pa, batch, z1, g, 0, N);

    // ---- layer 2: xw = z1 @ W2 ; agg (in d_out) ; z2 = prelu(agg + b2) in place
    gemm128_wmma_kernel<<<gemm_blocks, 256, 0, stream>>>(z1, W2, xw, N);
    init_agg_kernel<<<cdiv(nchunks, B), B, 0, stream>>>(dis, xw, agg, N);
    scatter_kernel<<<cdiv(E, 8), 256, 0, stream>>>(erow, ecol, norm, xw, agg, E);
    finalize_kernel<<<cdiv(nchunks, B), B, 0, stream>>>(agg, b2, pa, batch, z2, g, HID, N);
}